// ChannelAttention_wn_2284922602140
// MI455X (gfx1250) — compile-verified
//
#include <hip/hip_runtime.h>
#include <stdint.h>
#include <math.h>

// ---------------------------------------------------------------------------
// ChannelAttention on MI455X (gfx1250), compile-only tuned.
// x: [8, 64, 32, 128, 128] fp32.  Bandwidth-bound: 1 GiB in, 512 floats out.
// Strategy: TDM-stage (64ch x 128sites) tiles to LDS (read global ONCE),
// two LDS passes (norm, then accumulate), WMMA f32 16x16x4 ones-trick for the
// block-level sum reduction, float atomics into a tiny workspace, tiny
// finalize kernel for relu/sigmoid.
// ---------------------------------------------------------------------------

typedef __attribute__((ext_vector_type(2))) float        v2f;
typedef __attribute__((ext_vector_type(8))) float        v8f;
typedef __attribute__((ext_vector_type(4))) unsigned int u32x4;
typedef __attribute__((ext_vector_type(8))) int          i32x8;
typedef __attribute__((ext_vector_type(4))) int          i32x4;

#define NCH       64                     // channels
#define SBLK      128                    // spatial sites per tile
#define TPW       8                      // tiles per workgroup
#define NTHREADS  128                    // 4 waves (wave32)
#define S_TOTAL   (32*128*128)           // 524288 spatial sites per (b,c)
#define BPB       (S_TOTAL/(SBLK*TPW))   // 512 blocks per batch
#define NBATCH    8

#if __has_builtin(__builtin_amdgcn_tensor_load_to_lds)
#define HAVE_TDM 1
#else
#define HAVE_TDM 0
#endif

#if __has_builtin(__builtin_amdgcn_wmma_f32_16x16x4_f32)
#define HAVE_WMMA_F32X4 1
#else
#define HAVE_WMMA_F32X4 0
#endif

__device__ __forceinline__ void atomicMaxFloat(float* addr, float val) {
  // Works for mixed signs given init bits 0xFF800000 (-inf).
  if (val >= 0.0f) atomicMax((int*)addr, __float_as_int(val));
  else             atomicMin((unsigned int*)addr, __float_as_uint(val));
}

__global__ void ca_init(float* ws_sum, unsigned int* ws_max_bits) {
  int i = blockIdx.x * blockDim.x + threadIdx.x;
  if (i < NBATCH * NCH) {
    ws_sum[i]      = 0.0f;
    ws_max_bits[i] = 0xFF800000u;  // -inf
  }
}

__global__ __launch_bounds__(NTHREADS) void ca_main(const float* __restrict__ x,
                                                    float* __restrict__ ws_sum,
                                                    float* __restrict__ ws_max) {
  // Double-buffered tile: [2][64 ch][128 sites] = 64 KiB LDS.
  __shared__ float buf[2][NCH * SBLK];

  const int    t          = threadIdx.x;
  const int    b          = blockIdx.x / BPB;
  const int    chunk      = blockIdx.x % BPB;
  const size_t batch_base = (size_t)b * NCH * S_TOTAL;

  float sumAcc[NCH], maxAcc[NCH];
#pragma unroll
  for (int c = 0; c < NCH; ++c) { sumAcc[c] = 0.0f; maxAcc[c] = -__builtin_inff(); }

#if HAVE_TDM
  // Build + issue a 2D TDM descriptor: tile = SBLK(x) x NCH(y) fp32, row
  // stride S_TOTAL, landing contiguously as [ch][site] in LDS buf[j&1].
  auto tdm_issue = [&](int j) {
    const size_t       s0  = (size_t)(chunk * TPW + j) * SBLK;
    unsigned long long ga  = (unsigned long long)(uintptr_t)(x + batch_base + s0);
    unsigned int       lds = (unsigned int)(uintptr_t)(&buf[j & 1][0]); // low 32b = LDS offset

    u32x4 g0;
    g0[0] = 1u;                                                   // count=1, gather off
    g0[1] = lds;                                                  // lds_addr
    g0[2] = (unsigned int)(ga & 0xFFFFFFFFull);                   // global_addr[31:0]
    g0[3] = (unsigned int)((ga >> 32) & 0x01FFFFFFull) | (2u << 30); // addr[56:32], type=2

    const unsigned long long dim0 = S_TOTAL, str0 = S_TOTAL;
    const unsigned int dim1 = NCH, tile0 = SBLK, tile1 = NCH;
    i32x8 g1;
    g1[0] = (int)(2u << 16);                                           // data_size=4B
    g1[1] = (int)(((unsigned)(dim0 & 0xFFFFull)) << 16);               // dim0[15:0]
    g1[2] = (int)(((unsigned)((dim0 >> 16) & 0xFFFFull)) | ((dim1 & 0xFFFFu) << 16));
    g1[3] = (int)(((dim1 >> 16) & 0xFFFFu) | (tile0 << 16));           // tile_dim0
    g1[4] = (int)(tile1);                                              // tile_dim1, tile_dim2=0
    g1[5] = (int)(str0 & 0xFFFFFFFFull);                               // stride0[31:0]
    g1[6] = (int)((str0 >> 32) & 0xFFFFull);                           // stride0[47:32], stride1=0
    g1[7] = 0;
    i32x4 g2 = {0, 0, 0, 0};                 // D# group 2 (unused: tile_dim3=0)
    i32x4 g3 = {0, 0, 0, 0};                 // D# group 3 (unused: tile_dim4=0)
    i32x8 g4 = {0, 0, 0, 0, 0, 0, 0, 0};     // extra operand on clang-23 builtin
    __builtin_amdgcn_tensor_load_to_lds(g0, g1, g2, g3, g4, 0);
  };
#else
  auto coop_load = [&](int j) {
    const size_t s0 = (size_t)(chunk * TPW + j) * SBLK;
    for (int idx = t; idx < NCH * SBLK; idx += NTHREADS) {
      int c = idx >> 7, r = idx & (SBLK - 1);
      buf[j & 1][idx] = x[batch_base + (size_t)c * S_TOTAL + s0 + r];
    }
  };
#endif

  const bool wave0 = (t < 32);
#if HAVE_TDM
  if (wave0) tdm_issue(0);
#endif

  for (int j = 0; j < TPW; ++j) {
#if HAVE_TDM
    if (wave0) {
      if (j + 1 < TPW) { tdm_issue(j + 1); __builtin_amdgcn_s_wait_tensorcnt(1); }
      else             { __builtin_amdgcn_s_wait_tensorcnt(0); }
    }
    __syncthreads();                 // tile j visible to all waves
#else
    coop_load(j);
    __syncthreads();
#endif
    const float* tile = &buf[j & 1][0];

    // Pass 1: sum of squares over channels for site t (conflict-free LDS).
    float ss = 0.0f;
#pragma unroll
    for (int c = 0; c < NCH; ++c) { float v = tile[c * SBLK + t]; ss = fmaf(v, v, ss); }
    float inv = 1.0f / fmaxf(sqrtf(ss), 1e-12f);

    // Pass 2: normalize and fold into per-channel accumulators.
#pragma unroll
    for (int c = 0; c < NCH; ++c) {
      float v = tile[c * SBLK + t] * inv;
      sumAcc[c] += v;
      maxAcc[c] = fmaxf(maxAcc[c], v);
    }
    __syncthreads();                 // buffer safe to overwrite
  }

  // ---- Block-level reduction: partials P/Q are [128 threads][64 ch] in LDS.
  float* P = &buf[0][0];             // sums  (8192 floats)
  float* Q = P + NTHREADS * NCH;     // maxes (8192 floats)
#pragma unroll
  for (int c = 0; c < NCH; ++c) { P[t * NCH + c] = sumAcc[c]; Q[t * NCH + c] = maxAcc[c]; }
  __syncthreads();

  float* wsS = ws_sum + b * NCH;
  float* wsM = ws_max + b * NCH;

#if HAVE_WMMA_F32X4
  // Column-sums of P via ones-matrix WMMA: D = 1(16x4) x B(4x16) + C.
  // Sum over K is invariant to the exact B lane mapping, so this is robust.
  if (t < 32) {                      // wave 0, EXEC all ones for WMMA
    const int hv = t >> 4;           // lane half
    const int n  = t & 15;           // column within 16-channel block
    v2f ones; ones[0] = 1.0f; ones[1] = 1.0f;
#pragma unroll
    for (int nb = 0; nb < 4; ++nb) {
      v8f acc = {0.f, 0.f, 0.f, 0.f, 0.f, 0.f, 0.f, 0.f};
#pragma unroll 4
      for (int k0 = 0; k0 < NTHREADS; k0 += 4) {
        v2f bv;
        bv[0] = P[(k0 + hv)     * NCH + nb * 16 + n];
        bv[1] = P[(k0 + 2 + hv) * NCH + nb * 16 + n];
        acc = __builtin_amdgcn_wmma_f32_16x16x4_f32(false, ones, false, bv,
                                                    (short)0, acc, false, false);
      }
      if (t < 16) atomicAdd(&wsS[nb * 16 + t], acc[0]);  // D[0][n] lives in lane n, vgpr0
    }
  }
#else
  if (t < NCH) {
    float s = 0.0f;
    for (int r = 0; r < NTHREADS; ++r) s += P[r * NCH + t];
    atomicAdd(&wsS[t], s);
  }
#endif

  if (t < NCH) {
    float m = -__builtin_inff();
    for (int r = 0; r < NTHREADS; ++r) m = fmaxf(m, Q[r * NCH + t]);
    atomicMaxFloat(&wsM[t], m);
  }
}

__global__ void ca_final(const float* __restrict__ ws_sum,
                         const float* __restrict__ ws_max,
                         float* __restrict__ out) {
  int i = blockIdx.x * blockDim.x + threadIdx.x;
  if (i < NBATCH * NCH) {
    float o = ws_sum[i] * (1.0f / (float)S_TOTAL) + ws_max[i];
    o = fmaxf(o, 0.0f);                 // relu(avg + max)
    float z = o * o;                    // diagonal(out @ out^T)
    out[i] = 1.0f / (1.0f + expf(-z));  // sigmoid
  }
}

extern "C" void kernel_launch(void* const* d_in, const int* in_sizes, int n_in,
                              void* d_out, int out_size, void* d_ws, size_t ws_size,
                              hipStream_t stream) {
  const float* x      = (const float*)d_in[0];
  float*       out    = (float*)d_out;
  float*       ws_sum = (float*)d_ws;
  float*       ws_max = ws_sum + NBATCH * NCH;

  ca_init<<<2, 256, 0, stream>>>(ws_sum, (unsigned int*)ws_max);
  ca_main<<<NBATCH * BPB, NTHREADS, 0, stream>>>(x, ws_sum, ws_max);
  ca_final<<<2, 256, 0, stream>>>(ws_sum, ws_max, out);

  (void)in_sizes; (void)n_in; (void)out_size; (void)ws_size;
}